// EpisodicMemory_76355928588733
// MI455X (gfx1250) — compile-verified
//
#include <hip/hip_runtime.h>
#include <hip/hip_bf16.h>
#include <math.h>

// EpisodicMemory for MI455X (gfx1250, wave32).
//  * scores = (1/L) * <episode_n, v> + c with v = Wk.T @ (Wq q + bq):
//    one pure-bandwidth 256MB pass (the true roofline cost, ~11us @ 23.3 TB/s).
//  * BiLSTM GEMMs on V_WMMA_F32_16X16X4_F32 (fp32-exact matrix cores).
//  * w_ih panels and w_hh slices are LDS-resident (CDNA5: 320KB/WGP), so the
//    sequential LSTM recurrence does ZERO per-step weight re-reads from L2:
//    inner loop is ds_load_b64 -> v_wmma. h-state is exchanged between the 4
//    workgroups of a direction via a global atomic split-barrier per step.

#define NN    1024
#define LL    128
#define DD    512
#define HH    256
#define G4    1024          // 4*H
#define KSEL  5
#define MROWS (KSEL*LL)     // 640 rows (m = b*128+t), 40 M-tiles of 16

typedef float v2f __attribute__((ext_vector_type(2)));
typedef float v8f __attribute__((ext_vector_type(8)));

__device__ __forceinline__ float sigf(float x){ return 1.0f/(1.0f+expf(-x)); }

// ---------------- query projection: qp = Wq @ q + bq ----------------
__global__ void qproj_kernel(const float* __restrict__ Wq, const float* __restrict__ q,
                             const float* __restrict__ bq, float* __restrict__ qp){
  __shared__ float qs[DD];
  int tid = threadIdx.x;
  for (int j = tid; j < DD; j += 256) qs[j] = q[j];
  __syncthreads();
  int i = blockIdx.x*256 + tid;
  float acc = bq[i];
  const float4* row = (const float4*)(Wq + (size_t)i*DD);
  for (int j4 = 0; j4 < DD/4; ++j4){
    float4 wv = row[j4];
    int j = j4*4;
    acc += wv.x*qs[j] + wv.y*qs[j+1] + wv.z*qs[j+2] + wv.w*qs[j+3];
  }
  qp[i] = acc;
}

// ---------------- v = Wk.T @ qp ; c = bk . qp ----------------
__global__ void vproj_kernel(const float* __restrict__ Wk, const float* __restrict__ bk,
                             const float* __restrict__ qp, float* __restrict__ v,
                             float* __restrict__ cbias){
  __shared__ float qs[DD];
  __shared__ float red[256];
  int tid = threadIdx.x;
  for (int j = tid; j < DD; j += 256) qs[j] = qp[j];
  __syncthreads();
  if (blockIdx.x < 2){
    int d = blockIdx.x*256 + tid;
    float acc = 0.f;
    for (int i = 0; i < DD; ++i) acc += Wk[(size_t)i*DD + d]*qs[i];  // coalesced rows
    v[d] = acc;
  } else {
    float acc = 0.f;
    for (int i = tid; i < DD; i += 256) acc += bk[i]*qs[i];
    red[tid] = acc; __syncthreads();
    for (int off = 128; off > 0; off >>= 1){
      if (tid < off) red[tid] += red[tid+off];
      __syncthreads();
    }
    if (tid == 0) *cbias = red[0];
  }
}

// ---------------- scores[n] = (1/L) * <episode_n, v> + c  (256MB stream) ----------------
__global__ void scores_kernel(const float* __restrict__ ep, const float* __restrict__ v,
                              const float* __restrict__ cbias, float* __restrict__ scores){
  __shared__ float vs[DD];
  __shared__ float red[256];
  int tid = threadIdx.x;
  vs[tid] = v[tid]; vs[tid+256] = v[tid+256];
  __syncthreads();
  const float4* base = (const float4*)(ep + (size_t)blockIdx.x * (LL*DD));
  float acc = 0.f;
  for (int it = 0; it < (LL*DD/4)/256; ++it){   // 64 iterations of b128 loads
    int f4 = it*256 + tid;
    __builtin_prefetch(base + f4 + 2048, 0, 3); // speculative RT prefetch ahead
    float4 e = base[f4];
    int d = (f4 & 127) * 4;                     // column within D (row changes, v doesn't)
    acc += e.x*vs[d] + e.y*vs[d+1] + e.z*vs[d+2] + e.w*vs[d+3];
  }
  red[tid] = acc; __syncthreads();
  for (int off = 128; off > 0; off >>= 1){
    if (tid < off) red[tid] += red[tid+off];
    __syncthreads();
  }
  if (tid == 0) scores[blockIdx.x] = red[0]*(1.0f/LL) + *cbias;
}

// ---------------- top-k (k=5), w = 1/(1+age*0.01) ----------------
__global__ void topk_kernel(const float* __restrict__ scores, const float* __restrict__ ages,
                            int* __restrict__ topidx, float* __restrict__ topw){
  __shared__ float sv[NN];
  __shared__ float r[NN];
  __shared__ int   rid[NN];
  int tid = threadIdx.x;
  sv[tid] = scores[tid];
  __syncthreads();
  for (int it = 0; it < KSEL; ++it){
    r[tid] = sv[tid]; rid[tid] = tid;
    __syncthreads();
    for (int off = 512; off > 0; off >>= 1){
      if (tid < off){
        if (r[tid+off] > r[tid]){ r[tid] = r[tid+off]; rid[tid] = rid[tid+off]; }
      }
      __syncthreads();
    }
    if (tid == 0){
      int win = rid[0];
      topidx[it] = win;
      topw[it] = 1.0f/(1.0f + ages[win]*0.01f);
      sv[win] = -INFINITY;
    }
    __syncthreads();
  }
}

// ---------------- gather + age-weight scale; also zero barrier counters ----------------
__global__ void gather_kernel(const float* __restrict__ ep, const int* __restrict__ topidx,
                              const float* __restrict__ topw, float* __restrict__ x0,
                              int* __restrict__ bar){
  if (blockIdx.x == 0 && threadIdx.x < 4) bar[threadIdx.x] = 0;  // (layer,dir) barrier counters
  int kk = blockIdx.x >> 7;
  int l  = blockIdx.x & 127;
  float w = topw[kk];
  const float4* src = (const float4*)(ep + ((size_t)topidx[kk]*LL + l)*DD);
  float4* dst = (float4*)(x0 + ((size_t)kk*LL + l)*DD);
  float4 e = src[threadIdx.x];
  e.x *= w; e.y *= w; e.z *= w; e.w *= w;
  dst[threadIdx.x] = e;
}

// ---------------- input projection GEMM: pre = x @ w_ih.T + (b_ih + b_hh) ----------------
// grid = (16, 2): 16 N-tile-groups x 2 directions. Each block keeps its 64x512 B panel
// (128 KB) resident in LDS and loops over all 40 M-tiles, staging each 16x512 A tile
// (32 KB). B is read from L2 exactly once. Inner loop: ds_load_b64 x2 -> v_wmma.
__global__ __launch_bounds__(128) void inproj_kernel(const float* __restrict__ x,
        const float* __restrict__ wih_f, const float* __restrict__ wih_b,
        const float* __restrict__ bih_f, const float* __restrict__ bhh_f,
        const float* __restrict__ bih_b, const float* __restrict__ bhh_b,
        float* __restrict__ pre){
  __shared__ float Bs[64*DD];                 // 128 KB: rows n = ntg*64 .. +64
  __shared__ float As[16*DD];                 // 32 KB
  int dir = blockIdx.y;
  const float* wih = dir ? wih_b : wih_f;
  const float* bih = dir ? bih_b : bih_f;
  const float* bhh = dir ? bhh_b : bhh_f;
  float* predir = pre + (size_t)dir * (MROWS*(size_t)G4);
  int tid = threadIdx.x;
  int ntg = blockIdx.x;
  // stage B panel once (64 rows x 512), coalesced float4
  for (int it = 0; it < 64; ++it){
    int f4 = it*128 + tid;
    int lr = f4 >> 7;                         // 128 float4 per row
    int c4 = f4 & 127;
    *(float4*)&Bs[lr*DD + c4*4] = ((const float4*)(wih + (size_t)(ntg*64+lr)*DD))[c4];
  }
  int lane = tid & 31;
  int wave = tid >> 5;
  int klo  = (lane >> 4) << 1;                // lanes 0-15: K0,K1 ; lanes 16-31: K2,K3
  int mrow = lane & 15;
  int mhi  = (lane >> 4) << 3;                // C layout: lanes 16-31 hold M = r+8
  int nt   = ntg*4 + wave;
  int n    = nt*16 + mrow;
  float bias = bih[n] + bhh[n];
  const float* Bptr   = &Bs[(wave*16 + mrow)*DD];
  const float* As_row = &As[mrow*DD];
  for (int mt = 0; mt < MROWS/16; ++mt){
    __syncthreads();
    for (int it = 0; it < 16; ++it){
      int f4  = it*128 + tid;
      int row = f4 >> 7;
      int c4  = f4 & 127;
      *(float4*)&As[row*DD + c4*4] = ((const float4*)(x + (size_t)(mt*16+row)*DD))[c4];
    }
    __syncthreads();
    v8f acc = {};
    for (int k0 = 0; k0 < DD; k0 += 4){
      v2f a = *(const v2f*)(As_row + k0 + klo);
      v2f b = *(const v2f*)(Bptr + k0 + klo);
      acc = __builtin_amdgcn_wmma_f32_16x16x4_f32(false, a, false, b, (short)0, acc, false, false);
    }
    for (int r = 0; r < 8; ++r){
      int m = mt*16 + r + mhi;
      predir[(size_t)m*G4 + n] = acc[r] + bias;
    }
  }
}

// ---------------- LSTM recurrence: LDS-resident weights, 4 WGs per direction ----------------
// blockIdx.x = dir*4 + j. WG j owns hidden slice n in [j*64, j*64+64) for all 4 gates:
// 256 rows x 256 K of w_hh = 256 KB, staged in LDS once, reused for all 128 steps.
// Per step: G = Hs(16x256, padded rows zero) @ Bq.T via WMMA; fused gate update for the
// slice; h exchanged between the direction's 4 WGs through hglob + atomic split-barrier.
__global__ __launch_bounds__(512) void rec_kernel(const float* __restrict__ pre,
        const float* __restrict__ whh_f, const float* __restrict__ whh_b,
        float* __restrict__ out, float* __restrict__ hglob,
        int* __restrict__ bar, int layer){
  __shared__ float Bq[256*HH];               // 256 KB weight slice
  __shared__ float Hs[16*HH];                // 16 KB A operand (rows >=5 stay zero)
  __shared__ float Gs[KSEL*256];             // gate preacts for this slice
  __shared__ float Cs[KSEL*64];              // cell state slice
  int dir = blockIdx.x >> 2;
  int j   = blockIdx.x & 3;
  const float* whh    = dir ? whh_b : whh_f;
  const float* predir = pre + (size_t)dir*(MROWS*(size_t)G4);
  float* hg = hglob + (size_t)dir*(KSEL*HH);
  int* cnt  = bar + layer*2 + dir;
  int tid = threadIdx.x;
  // stage w_hh slice: local row lr = gate*64 + nl  <->  global row gate*256 + j*64 + nl
  for (int it = 0; it < 32; ++it){
    int f4 = it*512 + tid;
    int lr = f4 >> 6;                        // 64 float4 per 256-float row
    int c4 = f4 & 63;
    int grow = (lr >> 6)*HH + j*64 + (lr & 63);
    *(float4*)&Bq[lr*HH + c4*4] = ((const float4*)(whh + (size_t)grow*HH))[c4];
  }
  for (int i = tid; i < 16*HH;   i += 512) Hs[i] = 0.f;
  for (int i = tid; i < KSEL*64; i += 512) Cs[i] = 0.f;
  __syncthreads();
  int lane = tid & 31;
  int wave = tid >> 5;                       // 16 waves = 16 local N-tiles
  int klo  = (lane >> 4) << 1;
  int mrow = lane & 15;
  const float* Hrow = &Hs[mrow*HH];
  const float* Bptr = &Bq[(wave*16 + mrow)*HH];
  for (int s = 0; s < LL; ++s){
    int t = dir ? (LL-1-s) : s;              // backward dir scans reversed, writes at t
    v8f acc = {};
    for (int k0 = 0; k0 < HH; k0 += 4){
      v2f a = *(const v2f*)(Hrow + k0 + klo);
      v2f b = *(const v2f*)(Bptr + k0 + klo);
      acc = __builtin_amdgcn_wmma_f32_16x16x4_f32(false, a, false, b, (short)0, acc, false, false);
    }
    if (lane < 16){                          // real batch rows b<5 live in lanes 0-15
      int localn = wave*16 + lane;           // 0..255
      int gate = localn >> 6;
      int nl   = localn & 63;
      int ng   = gate*HH + j*64 + nl;        // global gate index 0..1023
      for (int r = 0; r < KSEL; ++r){
        Gs[r*256 + localn] = acc[r] + predir[(size_t)(r*LL + t)*G4 + ng];
      }
    }
    __syncthreads();
    if (tid < KSEL*64){
      int b  = tid >> 6;
      int nl = tid & 63;
      float gi = Gs[b*256 +       nl];
      float gf = Gs[b*256 +  64 + nl];
      float gc = Gs[b*256 + 128 + nl];
      float go = Gs[b*256 + 192 + nl];
      float cn = sigf(gf)*Cs[tid] + sigf(gi)*tanhf(gc);
      float h  = sigf(go)*tanhf(cn);
      Cs[tid] = cn;
      hg[b*HH + j*64 + nl] = h;
      out[(size_t)(b*LL + t)*DD + dir*HH + j*64 + nl] = h;
    }
    // split-barrier across the 4 WGs of this (layer,dir): monotonic counter
    __threadfence();
    __syncthreads();
    if (tid == 0){
      atomicAdd(cnt, 1);
      int target = 4*(s+1);
      while (__hip_atomic_load(cnt, __ATOMIC_ACQUIRE, __HIP_MEMORY_SCOPE_AGENT) < target){
        __builtin_amdgcn_s_sleep(1);
      }
    }
    __syncthreads();
    // refresh full h state (all 4 slices) for next step's A operand
    if (tid < KSEL*HH) Hs[(tid >> 8)*HH + (tid & 255)] = hg[tid];
    __syncthreads();
  }
}

// ---------------- attention + softmax + context ----------------
__global__ void attn_kernel(const float* __restrict__ cs, const float* __restrict__ y,
                            float* __restrict__ out){
  __shared__ float css[DD];
  __shared__ float att[LL];
  __shared__ float tmp;
  int b = blockIdx.x;
  int tid = threadIdx.x;
  css[tid] = cs[tid]; css[tid+256] = cs[tid+256];
  __syncthreads();
  int lane = tid & 31;
  int wave = tid >> 5;                       // 8 waves
  for (int t = wave; t < LL; t += 8){
    const float* yr = y + (size_t)(b*LL + t)*DD;
    float p = 0.f;
    for (int d = lane; d < DD; d += 32) p += css[d]*yr[d];
    for (int off = 16; off > 0; off >>= 1) p += __shfl_down(p, off, 32);
    if (lane == 0) att[t] = p;
  }
  __syncthreads();
  if (tid == 0){
    float mx = att[0];
    for (int t = 1; t < LL; ++t) mx = fmaxf(mx, att[t]);
    tmp = mx;
  }
  __syncthreads();
  if (tid < LL) att[tid] = expf(att[tid] - tmp);
  __syncthreads();
  if (tid == 0){
    float sm = 0.f;
    for (int t = 0; t < LL; ++t) sm += att[t];
    tmp = sm;
  }
  __syncthreads();
  if (tid < LL) att[tid] /= tmp;
  __syncthreads();
  for (int d = tid; d < DD; d += 256){
    float acc = 0.f;
    for (int t = 0; t < LL; ++t) acc += att[t]*y[(size_t)(b*LL + t)*DD + d];
    out[(size_t)b*DD + d] = acc;
  }
}

extern "C" void kernel_launch(void* const* d_in, const int* in_sizes, int n_in,
                              void* d_out, int out_size, void* d_ws, size_t ws_size,
                              hipStream_t stream){
  const float* episodes = (const float*)d_in[0];
  const float* query    = (const float*)d_in[1];
  const float* cstate   = (const float*)d_in[2];
  const float* ages     = (const float*)d_in[3];
  const float* Wq = (const float*)d_in[4];
  const float* bq = (const float*)d_in[5];
  const float* Wk = (const float*)d_in[6];
  const float* bk = (const float*)d_in[7];
  // order per setup_inputs: l0(8..11), l0r(12..15), l1(16..19), l1r(20..23)
  const float* w_ih[4] = {(const float*)d_in[8],  (const float*)d_in[12],
                          (const float*)d_in[16], (const float*)d_in[20]};
  const float* w_hh[4] = {(const float*)d_in[9],  (const float*)d_in[13],
                          (const float*)d_in[17], (const float*)d_in[21]};
  const float* b_ih[4] = {(const float*)d_in[10], (const float*)d_in[14],
                          (const float*)d_in[18], (const float*)d_in[22]};
  const float* b_hh[4] = {(const float*)d_in[11], (const float*)d_in[15],
                          (const float*)d_in[19], (const float*)d_in[23]};
  float* out = (float*)d_out;

  float* ws     = (float*)d_ws;
  float* qp     = ws;                         // 512
  float* v      = ws + 512;                   // 512
  float* cbias  = ws + 1024;                  // 1
  float* scores = ws + 1040;                  // 1024
  int*   topidx = (int*)(ws + 2064);          // 8
  float* topw   = ws + 2072;                  // 8
  float* x0     = ws + 2080;                  // 5*128*512 (16B aligned)
  float* x1     = x0 + (size_t)KSEL*LL*DD;    // 5*128*512
  float* y      = x1 + (size_t)KSEL*LL*DD;    // 5*128*512
  float* pre    = y  + (size_t)KSEL*LL*DD;    // 2 * 640*1024 per-direction preacts
  float* hglob  = pre + 2*(size_t)MROWS*G4;   // 2 * 5*256 h-state exchange
  int*   bar    = (int*)(hglob + 2*KSEL*HH);  // 4 barrier counters (layer,dir)

  qproj_kernel <<<2, 256, 0, stream>>>(Wq, query, bq, qp);
  vproj_kernel <<<3, 256, 0, stream>>>(Wk, bk, qp, v, cbias);
  scores_kernel<<<NN, 256, 0, stream>>>(episodes, v, cbias, scores);
  topk_kernel  <<<1, 1024, 0, stream>>>(scores, ages, topidx, topw);
  gather_kernel<<<KSEL*LL, 128, 0, stream>>>(episodes, topidx, topw, x0, bar);
  // layer 0 (fwd+bwd concurrent via grid dims)
  inproj_kernel<<<dim3(16,2), 128, 0, stream>>>(x0, w_ih[0], w_ih[1], b_ih[0], b_hh[0], b_ih[1], b_hh[1], pre);
  rec_kernel   <<<8, 512, 0, stream>>>(pre, w_hh[0], w_hh[1], x1, hglob, bar, 0);
  // layer 1
  inproj_kernel<<<dim3(16,2), 128, 0, stream>>>(x1, w_ih[2], w_ih[3], b_ih[2], b_hh[2], b_ih[3], b_hh[3], pre);
  rec_kernel   <<<8, 512, 0, stream>>>(pre, w_hh[2], w_hh[3], y, hglob, bar, 1);
  attn_kernel  <<<KSEL, 256, 0, stream>>>(cstate, y, out);
}